// ModelWithLoss_86792699118011
// MI455X (gfx1250) — compile-verified
//
#include <hip/hip_runtime.h>
#include <math.h>

// ---------------------------------------------------------------------------
// Problem constants (from reference setup_inputs)
// ---------------------------------------------------------------------------
#define B_IMG   32
#define N_ANCH  46035
#define M_BOX   64
#define SEG_C   3
#define SEG_H   384
#define SEG_W   640
#define HW      (SEG_H * SEG_W)          // 245760
#define SEG_N   (B_IMG * HW)             // 7,864,320 per channel

#define DET_BLK 64                       // blocks per image (detection)
#define SEG_BLK 512                      // blocks per channel (segmentation)

typedef __attribute__((ext_vector_type(2))) float v2f;
typedef __attribute__((ext_vector_type(8))) float v8f;

// ---------------------------------------------------------------------------
// Wave-level sum using V_WMMA_F32_16X16X4_F32:
//   A = ones(16x4), B row0/row1 = per-lane values (1 VGPR), rows2/3 = 0.
//   => D[m][n] = v[n] + v[n+16]  (fp32, RNE — no precision loss)
//   D VGPR0: lanes 0-15 hold (M=0, N=lane); lanes 16-31 hold (M=8, N=lane-16),
//   which is the same value, so 4 xor-shuffles within each 16-lane half
//   give every lane the full 32-lane sum.
// EXEC must be all ones at the call site (called only from converged code in
// fully-populated 256-thread blocks).
// ---------------------------------------------------------------------------
__device__ __forceinline__ float wave_sum(float v) {
#if defined(__gfx1250__) && __has_builtin(__builtin_amdgcn_wmma_f32_16x16x4_f32)
  v2f a;  a[0] = 1.0f; a[1] = 1.0f;     // A = all ones (both K-pairs)
  v2f bm; bm[0] = v;   bm[1] = 0.0f;    // B rows 0-1 = lane values, rows 2-3 = 0
  v8f c = {};
  c = __builtin_amdgcn_wmma_f32_16x16x4_f32(
      /*neg_a=*/false, a, /*neg_b=*/false, bm,
      /*c_mod=*/(short)0, c, /*reuse_a=*/false, /*reuse_b=*/false);
  float s = c[0];                        // per-lane: v[n] + v[n+16]
#else
  float s = v;
  s += __shfl_xor(s, 16, 32);
#endif
  s += __shfl_xor(s, 1, 32);
  s += __shfl_xor(s, 2, 32);
  s += __shfl_xor(s, 4, 32);
  s += __shfl_xor(s, 8, 32);
  return s;
}

// ---------------------------------------------------------------------------
// Detection loss partials: one (cls_sum, reg_sum, num_pos) triple per block.
// grid = (DET_BLK, B_IMG), block = 256.
// ---------------------------------------------------------------------------
__global__ void det_loss_kernel(const float* __restrict__ cls,
                                const float* __restrict__ reg,
                                const float* __restrict__ anchors,
                                const float* __restrict__ annot,
                                float* __restrict__ partials) {
  const int b   = blockIdx.y;
  const int tid = threadIdx.x;

  __shared__ float4 sBox[M_BOX];   // x1,y1,x2,y2
  __shared__ float  sMeta[M_BOX];  // box area if valid, else -1
  __shared__ float  redc[8], redr[8], redn[8];

  if (tid < M_BOX) {
    const float* an = annot + ((size_t)b * M_BOX + tid) * 5;
    float x1 = an[0], y1 = an[1], x2 = an[2], y2 = an[3], lab = an[4];
    sBox[tid] = make_float4(x1, y1, x2, y2);
    float area = (x2 - x1) * (y2 - y1);
    sMeta[tid] = (lab != -1.0f) ? area : -1.0f;
  }
  __syncthreads();

  float cls_acc = 0.0f, reg_acc = 0.0f, np_acc = 0.0f;
  const float4* anc4 = (const float4*)anchors;
  const float4* reg4 = (const float4*)reg;

  for (int a = blockIdx.x * blockDim.x + tid; a < N_ANCH;
       a += gridDim.x * blockDim.x) {
    float4 an = anc4[a];                 // anchor: (y1, x1, y2, x2)
    float ay1 = an.x, ax1 = an.y, ay2 = an.z, ax2 = an.w;
    float aw = ax2 - ax1, ah = ay2 - ay1;
    float areaA = aw * ah;

    // IoU max / first-argmax over 64 boxes (box coords: x1,y1,x2,y2)
    float best = -1.0f;
    int bestm = 0;
#pragma unroll 4
    for (int m = 0; m < M_BOX; ++m) {
      float4 bx  = sBox[m];
      float meta = sMeta[m];
      float iw = fminf(ax2, bx.z) - fmaxf(ax1, bx.x);
      float ih = fminf(ay2, bx.w) - fmaxf(ay1, bx.y);
      iw = fmaxf(iw, 0.0f);
      ih = fmaxf(ih, 0.0f);
      float inter = iw * ih;
      float ua  = fmaxf(areaA + fmaxf(meta, 0.0f) - inter, 1e-8f);
      float iou = (meta >= 0.0f) ? (inter / ua) : -1.0f;
      if (iou > best) { best = iou; bestm = m; }   // strict > == first argmax
    }

    float4 gb   = sBox[bestm];
    float gArea = sMeta[bestm];                    // assigned is always valid
    bool  big   = gArea > 100.0f;
    bool  pos   = big ? (best >= 0.5f) : (best >= 0.15f);

    float p = cls[(size_t)b * N_ANCH + a];
    p = fminf(fmaxf(p, 1e-4f), 1.0f - 1e-4f);

    if (pos) {
      float omp = 1.0f - p;
      cls_acc += 0.25f * omp * omp * (-logf(p));
      np_acc  += 1.0f;

      float gw  = gb.z - gb.x, gh = gb.w - gb.y;
      float gcx = gb.x + 0.5f * gw;
      float gcy = gb.y + 0.5f * gh;
      gw = fmaxf(gw, 1.0f);
      gh = fmaxf(gh, 1.0f);
      float acx = ax1 + 0.5f * aw;
      float acy = ay1 + 0.5f * ah;
      float tdy = (gcy - acy) / ah;
      float tdx = (gcx - acx) / aw;
      float tdh = logf(gh / ah);
      float tdw = logf(gw / aw);

      float4 r = reg4[(size_t)b * N_ANCH + a];
      float d0 = fabsf(tdy - r.x);
      float d1 = fabsf(tdx - r.y);
      float d2 = fabsf(tdh - r.z);
      float d3 = fabsf(tdw - r.w);
      const float th = 1.0f / 9.0f, sub = 0.5f / 9.0f;
      reg_acc += (d0 <= th) ? 4.5f * d0 * d0 : d0 - sub;
      reg_acc += (d1 <= th) ? 4.5f * d1 * d1 : d1 - sub;
      reg_acc += (d2 <= th) ? 4.5f * d2 * d2 : d2 - sub;
      reg_acc += (d3 <= th) ? 4.5f * d3 * d3 : d3 - sub;
    } else {
      cls_acc += 0.75f * p * p * (-logf(1.0f - p));
    }
  }

  // converged here -> EXEC all ones for WMMA reduction
  float wc = wave_sum(cls_acc);
  float wr = wave_sum(reg_acc);
  float wn = wave_sum(np_acc);
  int lane = tid & 31, wid = tid >> 5;
  if (lane == 0) { redc[wid] = wc; redr[wid] = wr; redn[wid] = wn; }
  __syncthreads();
  if (tid == 0) {
    float c0 = 0.f, r0 = 0.f, n0 = 0.f;
    for (int w = 0; w < 8; ++w) { c0 += redc[w]; r0 += redr[w]; n0 += redn[w]; }
    float* o = partials + ((size_t)b * gridDim.x + blockIdx.x) * 3;
    o[0] = c0; o[1] = r0; o[2] = n0;
  }
}

// ---------------------------------------------------------------------------
// Segmentation partials: per (channel, block): S1=sum(x*y), S2=sum(x),
// S3=sum(y), F=focal sum.  grid = (SEG_BLK, SEG_C), block = 256.
// ---------------------------------------------------------------------------
__global__ void seg_loss_kernel(const float* __restrict__ seg,
                                const float* __restrict__ ann,
                                float* __restrict__ partials) {
  const int c   = blockIdx.y;
  const int tid = threadIdx.x;
  __shared__ float red[4][8];

  float s1 = 0.f, s2 = 0.f, s3 = 0.f, foc = 0.f;
  const unsigned step = gridDim.x * blockDim.x;        // 131072

  for (unsigned i = blockIdx.x * blockDim.x + tid; i < (unsigned)SEG_N;
       i += step) {
    unsigned bimg = i / (unsigned)HW;
    unsigned p    = i - bimg * (unsigned)HW;
    size_t idx = ((size_t)bimg * SEG_C + c) * HW + p;
    float x = seg[idx];
    float y = ann[idx];
    if (i + step < (unsigned)SEG_N) {                  // speculative stream hint
      __builtin_prefetch(seg + idx + step, 0, 0);
      __builtin_prefetch(ann + idx + step, 0, 0);
    }
    s1 = fmaf(x, y, s1);
    s2 += x;
    s3 += y;
    // stable BCE-with-logits: logaddexp(0,x) - x*y
    float logpt = fmaxf(x, 0.0f) - x * y + log1pf(expf(-fabsf(x)));
    float pt = expf(-logpt);
    float w  = 1.0f - pt;
    foc += w * w * logpt * (0.75f - 0.5f * y);         // alpha*y+(1-alpha)(1-y)
  }

  float r0 = wave_sum(s1);
  float r1 = wave_sum(s2);
  float r2 = wave_sum(s3);
  float r3 = wave_sum(foc);
  int lane = tid & 31, wid = tid >> 5;
  if (lane == 0) { red[0][wid] = r0; red[1][wid] = r1;
                   red[2][wid] = r2; red[3][wid] = r3; }
  __syncthreads();
  if (tid == 0) {
    float t0 = 0.f, t1 = 0.f, t2 = 0.f, t3 = 0.f;
    for (int w = 0; w < 8; ++w) {
      t0 += red[0][w]; t1 += red[1][w]; t2 += red[2][w]; t3 += red[3][w];
    }
    float* o = partials + ((size_t)c * gridDim.x + blockIdx.x) * 4;
    o[0] = t0; o[1] = t1; o[2] = t2; o[3] = t3;
  }
}

// ---------------------------------------------------------------------------
// Deterministic finalize: per-image normalize + batch means + tversky/focal.
// ---------------------------------------------------------------------------
__global__ void finalize_kernel(const float* __restrict__ detp,
                                const float* __restrict__ segp,
                                float* __restrict__ out) {
  const int tid = threadIdx.x;
  __shared__ float lcls[B_IMG], lreg[B_IMG];
  __shared__ float sseg[SEG_C][4];

  if (tid < B_IMG) {
    float c0 = 0.f, r0 = 0.f, n0 = 0.f;
    for (int k = 0; k < DET_BLK; ++k) {
      const float* p = detp + ((size_t)tid * DET_BLK + k) * 3;
      c0 += p[0]; r0 += p[1]; n0 += p[2];
    }
    float mnp = fmaxf(n0, 1.0f);
    lcls[tid] = c0 / mnp;
    lreg[tid] = (n0 > 0.0f) ? r0 / (4.0f * mnp) : 0.0f;
  }
  if (tid >= 32 && tid < 32 + SEG_C) {
    int c = tid - 32;
    float t0 = 0.f, t1 = 0.f, t2 = 0.f, t3 = 0.f;
    for (int k = 0; k < SEG_BLK; ++k) {
      const float* p = segp + ((size_t)c * SEG_BLK + k) * 4;
      t0 += p[0]; t1 += p[1]; t2 += p[2]; t3 += p[3];
    }
    sseg[c][0] = t0; sseg[c][1] = t1; sseg[c][2] = t2; sseg[c][3] = t3;
  }
  __syncthreads();

  if (tid == 0) {
    float cm = 0.f, rm = 0.f;
    for (int i = 0; i < B_IMG; ++i) { cm += lcls[i]; rm += lreg[i]; }
    cm /= (float)B_IMG;
    rm /= (float)B_IMG;
    out[0] = cm;
    out[1] = rm * 50.0f;

    float tv = 0.f, foct = 0.f;
    for (int c = 0; c < SEG_C; ++c) {
      float s1 = sseg[c][0], s2 = sseg[c][1], s3 = sseg[c][2];
      foct += sseg[c][3];
      // inter + alpha*fp + beta*fn with fp=S2-S1, fn=S3-S1
      float denom = fmaxf(s1 + 0.7f * (s2 - s1) + 0.3f * (s3 - s1), 1e-7f);
      float score = s1 / denom;
      float mask  = (s3 > 0.0f) ? 1.0f : 0.0f;
      tv += (1.0f - score) * mask;
    }
    tv /= (float)SEG_C;
    float tvp = powf(tv, 4.0f / 3.0f);
    float fm  = foct / (float)(B_IMG * SEG_C * HW);
    out[2] = tvp + fm;
  }
}

// ---------------------------------------------------------------------------
extern "C" void kernel_launch(void* const* d_in, const int* in_sizes, int n_in,
                              void* d_out, int out_size, void* d_ws,
                              size_t ws_size, hipStream_t stream) {
  (void)in_sizes; (void)n_in; (void)out_size; (void)ws_size;
  const float* cls    = (const float*)d_in[0];  // (B, A, 1)
  const float* reg    = (const float*)d_in[1];  // (B, A, 4)
  const float* anc    = (const float*)d_in[2];  // (1, A, 4)
  const float* annot  = (const float*)d_in[3];  // (B, 64, 5)
  const float* seg    = (const float*)d_in[4];  // (B, 3, H, W)
  const float* segann = (const float*)d_in[5];  // (B, 3, H, W)
  float* out = (float*)d_out;
  float* ws  = (float*)d_ws;

  float* detp = ws;                                   // 32*64*3 floats
  float* segp = ws + (size_t)B_IMG * DET_BLK * 3;     // 3*512*4 floats

  det_loss_kernel<<<dim3(DET_BLK, B_IMG), 256, 0, stream>>>(cls, reg, anc,
                                                            annot, detp);
  seg_loss_kernel<<<dim3(SEG_BLK, SEG_C), 256, 0, stream>>>(seg, segann, segp);
  finalize_kernel<<<1, 64, 0, stream>>>(detp, segp, out);
}